// LogitGenerator_30193620091605
// MI455X (gfx1250) — compile-verified
//
#include <hip/hip_runtime.h>
#include <math.h>

// ---------------------------------------------------------------------------
// LogitGenerator for MI455X (gfx1250), wave32, fp32 WMMA path.
//
// out[t, 0]      = sims[t, label[t]]                  (unmasked "pos")
// out[t, 1 + c]  = eq[label[t], c] ? -inf : sims[t,c] ("negs")
// sims = rownorm(x @ W^T + b) @ rownorm(E)^T / 0.1
//
// B matrices are stored "pair-packed": element (k, n) lives at
//   [(k>>1) * 2*N + 2*n + (k&1)]
// so a lane's B fragment (K=k0+2*lhalf and K+1 at fixed n) is one b64 load,
// and 16 lanes cover one contiguous 128B line.
// ---------------------------------------------------------------------------

typedef __attribute__((ext_vector_type(2))) float v2f;
typedef __attribute__((ext_vector_type(8))) float v8f;

#define T_ROWS   32768
#define D_ENC    768
#define D_FIN    256
#define C_CLS    504
#define C_PAD    512
#define OUT_W    505          // 1 + C_CLS
#define NORM_EPS 1e-8f

// ---------------- prep kernels (tiny) --------------------------------------

// WT2 pair-packed from W [256, 768]: (k, n) -> WT2[(k>>1)*512 + 2n + (k&1)]
__global__ void transpose_w_kernel(const float* __restrict__ W,
                                   float* __restrict__ WT2) {
  int idx = blockIdx.x * 256 + threadIdx.x;     // 768*256 total
  int k = idx >> 8;                              // 0..767
  int n = idx & 255;                             // 0..255
  WT2[(k >> 1) * (2 * D_FIN) + 2 * n + (k & 1)] = W[n * D_ENC + k];
}

// enT2 pair-packed normalized embeddings: (k, c) -> enT2[(k>>1)*1024+2c+(k&1)]
// columns c in [504,512) are zero padding.
__global__ void norm_emb_kernel(const float* __restrict__ E,
                                float* __restrict__ enT2) {
  int c = blockIdx.x;       // 0..511
  int t = threadIdx.x;      // 0..255 == k
  __shared__ float red[256];
  __shared__ float inv;
  float v = (c < C_CLS) ? E[c * D_FIN + t] : 0.0f;
  red[t] = v * v;
  __syncthreads();
  for (int s = 128; s > 0; s >>= 1) {
    if (t < s) red[t] += red[t + s];
    __syncthreads();
  }
  if (t == 0) inv = 1.0f / fmaxf(sqrtf(red[0]), NORM_EPS);
  __syncthreads();
  enT2[(t >> 1) * (2 * C_PAD) + 2 * c + (t & 1)] = (c < C_CLS) ? v * inv : 0.0f;
}

// eq[c1, c2] = all(E[c1,:] == E[c2,:]); stored with stride C_PAD
__global__ void eq_kernel(const float* __restrict__ E,
                          unsigned char* __restrict__ eq) {
  int c1 = blockIdx.x;      // 0..503
  int c2 = threadIdx.x;     // 0..511
  unsigned char r = 0;
  if (c2 < C_CLS) {
    const float4* a = (const float4*)(E + c1 * D_FIN);
    const float4* b = (const float4*)(E + c2 * D_FIN);
    bool same = true;
    #pragma unroll 4
    for (int d = 0; d < D_FIN / 4; ++d) {
      float4 av = a[d], bv = b[d];
      same = same && (av.x == bv.x) && (av.y == bv.y) &&
                     (av.z == bv.z) && (av.w == bv.w);
    }
    r = same ? 1 : 0;
  }
  eq[c1 * C_PAD + c2] = r;
}

// ---------------- kernel 1: projection GEMM + fused row-normalize ----------
// Block: 256 threads = 8 waves. Block tile: 16 rows x 256 cols.
// Wave w owns columns [32w, 32w+32) as two 16x16 WMMA tiles; K loop = 192.
__global__ void __launch_bounds__(256)
proj_norm_kernel(const float* __restrict__ x, const float* __restrict__ WT2,
                 const float* __restrict__ bproj, float* __restrict__ xn) {
  const int m0    = blockIdx.x * 16;
  const int tid   = threadIdx.x;
  const int wave  = tid >> 5;
  const int lane  = tid & 31;
  const int lhalf = lane >> 4;     // 0 | 1 (lane half)
  const int lmod  = lane & 15;
  const int n0    = wave * 32;

  __shared__ float part[8][16];    // per-wave partial sum-of-squares per row
  __shared__ float rowinv[16];

  v8f acc0 = {}; v8f acc1 = {};
  // A f32 16x4 layout: VGPR0 = K (k0 + 2*lhalf), VGPR1 = K+1, M = lmod
  const float* xrow  = x + (m0 + lmod) * D_ENC + 2 * lhalf;
  // B base: pair-packed row (k0/2 + lhalf), column n0+lmod
  const float* wbase = WT2 + lhalf * (2 * D_FIN) + 2 * (n0 + lmod);
  for (int k0 = 0; k0 < D_ENC; k0 += 4) {
    v2f a = *(const v2f*)(xrow + k0);                  // one b64
    const v2f* wp = (const v2f*)(wbase + (k0 >> 1) * (2 * D_FIN));
    v2f b0 = wp[0];                                     // cols n0+lmod
    v2f b1 = wp[16];                                    // cols n0+16+lmod
    acc0 = __builtin_amdgcn_wmma_f32_16x16x4_f32(false, a, false, b0,
                                                 (short)0, acc0, false, false);
    acc1 = __builtin_amdgcn_wmma_f32_16x16x4_f32(false, a, false, b1,
                                                 (short)0, acc1, false, false);
  }

  const float bias0 = bproj[n0 + lmod];
  const float bias1 = bproj[n0 + 16 + lmod];
  float sq[8];
  #pragma unroll
  for (int r = 0; r < 8; ++r) {
    acc0[r] += bias0;
    acc1[r] += bias1;
    sq[r] = acc0[r] * acc0[r] + acc1[r] * acc1[r];
  }
  // Deterministic reduction: shuffle within each 16-lane half, then LDS.
  #pragma unroll
  for (int r = 0; r < 8; ++r) {
    float s = sq[r];
    s += __shfl_xor(s, 1, 32);
    s += __shfl_xor(s, 2, 32);
    s += __shfl_xor(s, 4, 32);
    s += __shfl_xor(s, 8, 32);
    if (lmod == 0) part[wave][8 * lhalf + r] = s;   // row M = r + 8*lhalf
  }
  __syncthreads();
  if (tid < 16) {
    float s = 0.0f;
    #pragma unroll
    for (int w = 0; w < 8; ++w) s += part[w][tid];  // fixed order -> stable
    rowinv[tid] = 1.0f / fmaxf(sqrtf(s), NORM_EPS);
  }
  __syncthreads();
  // C/D layout: VGPR r, M = r + 8*lhalf, N = lmod
  #pragma unroll
  for (int r = 0; r < 8; ++r) {
    const int M = r + 8 * lhalf;
    const float inv = rowinv[M];
    xn[(m0 + M) * D_FIN + n0 + lmod]      = acc0[r] * inv;
    xn[(m0 + M) * D_FIN + n0 + 16 + lmod] = acc1[r] * inv;
  }
}

// ---------------- kernel 2: similarity GEMM + mask + pos -------------------
// Grid (2048, 2); 8 waves/block, two 16x16 tiles per wave (A reused x2);
// K loop = 64, 2 WMMAs per iteration.
__global__ void __launch_bounds__(256)
sims_kernel(const float* __restrict__ xn, const float* __restrict__ enT2,
            const unsigned char* __restrict__ eq, const int* __restrict__ label,
            float* __restrict__ out) {
  const int m0    = blockIdx.x * 16;
  const int tid   = threadIdx.x;
  const int wave  = tid >> 5;
  const int lane  = tid & 31;
  const int lhalf = lane >> 4;
  const int lmod  = lane & 15;
  const int n0    = blockIdx.y * 256 + wave * 32;   // 0..480 step 32

  __shared__ int lab[16];
  if (tid < 16) lab[tid] = label[m0 + tid];
  __syncthreads();

  v8f acc0 = {}; v8f acc1 = {};
  const float* xrow  = xn + (m0 + lmod) * D_FIN + 2 * lhalf;
  const float* ebase = enT2 + lhalf * (2 * C_PAD) + 2 * (n0 + lmod);
  for (int k0 = 0; k0 < D_FIN; k0 += 4) {
    v2f a = *(const v2f*)(xrow + k0);                  // one b64
    const v2f* ep = (const v2f*)(ebase + (k0 >> 1) * (2 * C_PAD));
    v2f b0 = ep[0];                                     // cols n0+lmod
    v2f b1 = ep[16];                                    // cols n0+16+lmod
    acc0 = __builtin_amdgcn_wmma_f32_16x16x4_f32(false, a, false, b0,
                                                 (short)0, acc0, false, false);
    acc1 = __builtin_amdgcn_wmma_f32_16x16x4_f32(false, a, false, b1,
                                                 (short)0, acc1, false, false);
  }

  const int   col0    = n0 + lmod;
  const int   col1    = n0 + 16 + lmod;
  const float NEG_INF = __int_as_float(0xFF800000);
  #pragma unroll
  for (int r = 0; r < 8; ++r) {
    const int M   = r + 8 * lhalf;
    const int row = m0 + M;
    const int lb  = lab[M];
    const float v0 = acc0[r] / 0.1f;                // matches reference / 0.1
    const float v1 = acc1[r] / 0.1f;
    if (col0 < C_CLS) {
      const bool msk = eq[lb * C_PAD + col0] != 0;
      out[row * OUT_W + 1 + col0] = msk ? NEG_INF : v0;
      if (col0 == lb) out[row * OUT_W] = v0;        // unmasked positive logit
    }
    if (col1 < C_CLS) {
      const bool msk = eq[lb * C_PAD + col1] != 0;
      out[row * OUT_W + 1 + col1] = msk ? NEG_INF : v1;
      if (col1 == lb) out[row * OUT_W] = v1;
    }
  }
}

// ---------------- launcher -------------------------------------------------
extern "C" void kernel_launch(void* const* d_in, const int* in_sizes, int n_in,
                              void* d_out, int out_size, void* d_ws,
                              size_t ws_size, hipStream_t stream) {
  (void)in_sizes; (void)n_in; (void)out_size; (void)ws_size;
  const float* x     = (const float*)d_in[0];   // [32768, 768]
  const int*   label = (const int*)d_in[1];     // [32768]
  const float* W     = (const float*)d_in[2];   // [256, 768]
  const float* bprj  = (const float*)d_in[3];   // [256]
  const float* E     = (const float*)d_in[4];   // [504, 256]
  float* out = (float*)d_out;                   // [32768, 505]

  char* ws = (char*)d_ws;
  float* xn   = (float*)ws;                                 // 32768*256*4
  float* enT2 = (float*)(ws + (size_t)T_ROWS * D_FIN * 4);  // 256*512*4
  float* WT2  = (float*)(ws + (size_t)T_ROWS * D_FIN * 4
                            + (size_t)D_FIN * C_PAD * 4);   // 768*256*4
  unsigned char* eqm = (unsigned char*)(ws + (size_t)T_ROWS * D_FIN * 4
                                           + (size_t)D_FIN * C_PAD * 4
                                           + (size_t)D_ENC * D_FIN * 4);

  transpose_w_kernel<<<D_ENC, 256, 0, stream>>>(W, WT2);
  norm_emb_kernel<<<C_PAD, 256, 0, stream>>>(E, enT2);
  eq_kernel<<<C_CLS, C_PAD, 0, stream>>>(E, eqm);
  proj_norm_kernel<<<T_ROWS / 16, 256, 0, stream>>>(x, WT2, bprj, xn);
  sims_kernel<<<dim3(T_ROWS / 16, 2), 256, 0, stream>>>(xn, enT2, eqm, label, out);
}